// PhysicsInformedWNTXentLoss2D_22084721836289
// MI455X (gfx1250) — compile-verified
//
#include <hip/hip_runtime.h>
#include <math.h>

#define NN    256
#define DDIM  512
#define TWOB  256

static constexpr float kTemp  = 0.1f;
static constexpr float kLam   = 0.1f;
static constexpr float kTwoPi = 6.283185307179586f;
static constexpr float kFourPi2 = 39.478417604357434f;

typedef float v2f __attribute__((ext_vector_type(2)));
typedef float v8f __attribute__((ext_vector_type(8)));

__device__ __forceinline__ v8f wmma4(v8f c, v2f a, v2f b) {
  return __builtin_amdgcn_wmma_f32_16x16x4_f32(false, a, false, b, (short)0, c, false, false);
}

// ---------------- workspace layout (float offsets) ----------------
constexpr size_t SZ_T     = 256ull * 256;          // twiddle table
constexpr size_t SZ_W     = 256ull * 144;          // weighted irfft table
constexpr size_t SZ_PLANE = 64ull * 256 * 256;     // 64 batch planes
constexpr size_t SZ_HALFT = 64ull * 144 * 256;     // 64 half-spectra (padded to 144)
constexpr size_t OFF_TC   = 0;
constexpr size_t OFF_TS   = OFF_TC + SZ_T;
constexpr size_t OFF_WC   = OFF_TS + SZ_T;
constexpr size_t OFF_WS2  = OFF_WC + SZ_W;
constexpr size_t OFF_UT   = OFF_WS2 + SZ_W;        // transposed u planes
constexpr size_t OFF_WR   = OFF_UT + SZ_PLANE;     // stage1 out / stage3 Pq re (reuse)
constexpr size_t OFF_WI   = OFF_WR + SZ_PLANE;     // stage1 out / stage3 Pq im (reuse)
constexpr size_t OFF_QHRT = OFF_WI + SZ_PLANE;     // transposed spectra [j][i]
constexpr size_t OFF_QHIT = OFF_QHRT + SZ_HALFT;
constexpr size_t OFF_VHRT = OFF_QHIT + SZ_HALFT;
constexpr size_t OFF_VHIT = OFF_VHRT + SZ_HALFT;
constexpr size_t OFF_PQR  = OFF_WR;                // reuse (stage1 data dead after stage2)
constexpr size_t OFF_PQI  = OFF_WI;
constexpr size_t OFF_PVR  = OFF_VHIT + SZ_HALFT;
constexpr size_t OFF_PVI  = OFF_PVR + SZ_HALFT;
constexpr size_t OFF_QF   = OFF_PVI + SZ_HALFT;    // q field
constexpr size_t OFF_VF   = OFF_QF + SZ_PLANE;     // v field
constexpr size_t OFF_RN   = OFF_VF + SZ_PLANE;     // 256x512
constexpr size_t OFF_SMAT = OFF_RN + 256ull * 512; // 256x256
constexpr size_t OFF_XI   = OFF_SMAT + 256ull * 256;
constexpr size_t OFF_NNS  = OFF_XI + 512;
constexpr size_t OFF_NSS  = OFF_NNS + 512;
constexpr size_t OFF_RL   = OFF_NSS + 1024;

// ---------------- twiddle tables ----------------
__global__ void k_init_tw(float* __restrict__ ws) {
  const int r = blockIdx.x, c = threadIdx.x;       // 256 x 256
  const int p = (r * c) & 255;
  const float th = kTwoPi * (float)p / 256.0f;
  ws[OFF_TC + (size_t)r * 256 + c] = cosf(th);
  ws[OFF_TS + (size_t)r * 256 + c] = sinf(th);
  if (c < 144) {                                   // weighted irfft-x table: r = x, c = j
    float wc = 0.f, wsn = 0.f;
    if (c <= 128) {
      float alpha = (c == 0 || c == 128) ? 1.f : 2.f;
      wc  = alpha * cosf(th) / 256.0f;
      wsn = alpha * sinf(th) / 256.0f;
    }
    ws[OFF_WC  + (size_t)r * 144 + c] = wc;
    ws[OFF_WS2 + (size_t)r * 144 + c] = wsn;
  }
}

// ---------------- transpose u0 planes (NS half) ----------------
__global__ void k_tr(const float* __restrict__ u, float* __restrict__ ws) {
  __shared__ float tile[32][33];
  const int b = blockIdx.z;
  const int x0 = blockIdx.x * 32, y0 = blockIdx.y * 32;
  const int tx = threadIdx.x, ty = threadIdx.y;    // (32,8)
  const float* up = u + (size_t)(64 + b) * NN * NN * 2;
  for (int i = 0; i < 32; i += 8)
    tile[ty + i][tx] = up[((size_t)(y0 + ty + i) * NN + (x0 + tx)) * 2];
  __syncthreads();
  float* Ut = ws + OFF_UT + (size_t)b * NN * NN;
  for (int i = 0; i < 32; i += 8)
    Ut[(size_t)(x0 + ty + i) * NN + (y0 + tx)] = tile[tx][ty + i];
}

// ---------------- stage 1: y-DFT as GEMM: W = T * Ut^T ----------------
__global__ void __launch_bounds__(32) k_s1(float* __restrict__ ws) {
  const int lane = threadIdx.x, lm = lane & 15, hi = lane >> 4;
  const int m0 = blockIdx.x * 16;                  // k rows
  const int n0 = blockIdx.y * 16;                  // x cols
  const int b  = blockIdx.z;
  const float* Ac = ws + OFF_TC + (size_t)(m0 + lm) * 256 + 2 * hi;
  const float* As = ws + OFF_TS + (size_t)(m0 + lm) * 256 + 2 * hi;
  const float* Bu = ws + OFF_UT + (size_t)b * NN * NN + (size_t)(n0 + lm) * 256 + 2 * hi;
  v8f c0 = {0.f,0.f,0.f,0.f,0.f,0.f,0.f,0.f}, c1 = c0;
  for (int k = 0; k < 256; k += 4) {
    v2f bu = *(const v2f*)(Bu + k);
    v2f ac = *(const v2f*)(Ac + k);
    v2f as = *(const v2f*)(As + k);
    c0 = wmma4(c0, ac, bu);
    c1 = wmma4(c1, as, bu);
  }
  float* Wr = ws + OFF_WR + (size_t)b * NN * NN;
  float* Wi = ws + OFF_WI + (size_t)b * NN * NN;
  const int mrow = m0 + 8 * hi;
  for (int r = 0; r < 8; ++r) {
    Wr[(size_t)(mrow + r) * 256 + n0 + lm] =  c0[r];
    Wi[(size_t)(mrow + r) * 256 + n0 + lm] = -c1[r];
  }
}

// ---------------- stage 2: x-DFT as GEMM + spectral multipliers (transposed out) ----------------
__global__ void __launch_bounds__(32) k_s2(float* __restrict__ ws) {
  const int lane = threadIdx.x, lm = lane & 15, hi = lane >> 4;
  const int m0 = blockIdx.x * 16;                  // j rows (0..143)
  const int n0 = blockIdx.y * 16;                  // i cols (0..255)
  const int b  = blockIdx.z;
  const float* Ac = ws + OFF_TC + (size_t)(m0 + lm) * 256 + 2 * hi;
  const float* As = ws + OFF_TS + (size_t)(m0 + lm) * 256 + 2 * hi;
  const float* Br = ws + OFF_WR + (size_t)b * NN * NN + (size_t)(n0 + lm) * 256 + 2 * hi;
  const float* Bi = ws + OFF_WI + (size_t)b * NN * NN + (size_t)(n0 + lm) * 256 + 2 * hi;
  v8f c0 = {0.f,0.f,0.f,0.f,0.f,0.f,0.f,0.f}, c1 = c0, c2 = c0, c3 = c0;
  for (int k = 0; k < 256; k += 4) {
    v2f ac = *(const v2f*)(Ac + k);
    v2f as = *(const v2f*)(As + k);
    v2f br = *(const v2f*)(Br + k);
    v2f bi = *(const v2f*)(Bi + k);
    c0 = wmma4(c0, ac, br);   // Tc * Wr
    c1 = wmma4(c1, as, bi);   // Ts * Wi
    c2 = wmma4(c2, ac, bi);   // Tc * Wi
    c3 = wmma4(c3, as, br);   // Ts * Wr
  }
  const int i = n0 + lm;
  const float kx = (i < 128) ? (float)i : (float)(i - 256);
  const int mrow = m0 + 8 * hi;
  const size_t base = (size_t)b * 144 * 256 + i;
  for (int r = 0; r < 8; ++r) {
    const int j = mrow + r;
    float qr = 0.f, qi = 0.f, vr = 0.f, vi = 0.f;
    if (j <= 128) {
      float ky = (j < 128) ? (float)j : -128.0f;
      float lap = kFourPi2 * (kx * kx + ky * ky);
      if (i == 0 && j == 0) lap = 1.0f;
      float ar = c0[r] + c1[r];
      float ai = c2[r] - c3[r];
      float pr = ar / lap, pim = ai / lap;
      qr = -kTwoPi * ky * pim;  qi =  kTwoPi * ky * pr;   // Qhat = (2*pi*i*ky)*psi
      vr =  kTwoPi * kx * pim;  vi = -kTwoPi * kx * pr;   // Vhat = (-2*pi*i*kx)*psi
    }
    const size_t o = base + (size_t)j * 256;
    ws[OFF_QHRT + o] = qr;  ws[OFF_QHIT + o] = qi;
    ws[OFF_VHRT + o] = vr;  ws[OFF_VHIT + o] = vi;
  }
}

// ---------------- stage 3: y-IFFT as GEMM (complex, both fields) ----------------
__global__ void __launch_bounds__(32) k_s3(float* __restrict__ ws) {
  const int lane = threadIdx.x, lm = lane & 15, hi = lane >> 4;
  const int m0 = blockIdx.x * 16;                  // y rows
  const int n0 = blockIdx.y * 16;                  // j cols (0..143)
  const int b  = blockIdx.z;
  const float* Ac = ws + OFF_TC + (size_t)(m0 + lm) * 256 + 2 * hi;
  const float* As = ws + OFF_TS + (size_t)(m0 + lm) * 256 + 2 * hi;
  const size_t hb = (size_t)b * 144 * 256 + (size_t)(n0 + lm) * 256 + 2 * hi;
  const float* Bqr = ws + OFF_QHRT + hb;
  const float* Bqi = ws + OFF_QHIT + hb;
  const float* Bvr = ws + OFF_VHRT + hb;
  const float* Bvi = ws + OFF_VHIT + hb;
  v8f z = {0.f,0.f,0.f,0.f,0.f,0.f,0.f,0.f};
  v8f qA = z, qB = z, qC = z, qD = z, vA = z, vB = z, vC = z, vD = z;
  for (int k = 0; k < 256; k += 4) {
    v2f ac = *(const v2f*)(Ac + k);
    v2f as = *(const v2f*)(As + k);
    v2f br = *(const v2f*)(Bqr + k);
    v2f bi = *(const v2f*)(Bqi + k);
    v2f cr = *(const v2f*)(Bvr + k);
    v2f ci = *(const v2f*)(Bvi + k);
    qA = wmma4(qA, ac, br);  qB = wmma4(qB, as, bi);
    qC = wmma4(qC, ac, bi);  qD = wmma4(qD, as, br);
    vA = wmma4(vA, ac, cr);  vB = wmma4(vB, as, ci);
    vC = wmma4(vC, ac, ci);  vD = wmma4(vD, as, cr);
  }
  const float inv = 1.0f / 256.0f;
  float* Pqr = ws + OFF_PQR + (size_t)b * 36864;
  float* Pqi = ws + OFF_PQI + (size_t)b * 36864;
  float* Pvr = ws + OFF_PVR + (size_t)b * 36864;
  float* Pvi = ws + OFF_PVI + (size_t)b * 36864;
  const int mrow = m0 + 8 * hi;
  for (int r = 0; r < 8; ++r) {
    const size_t o = (size_t)(mrow + r) * 144 + n0 + lm;
    Pqr[o] = (qA[r] - qB[r]) * inv;
    Pqi[o] = (qC[r] + qD[r]) * inv;
    Pvr[o] = (vA[r] - vB[r]) * inv;
    Pvi[o] = (vC[r] + vD[r]) * inv;
  }
}

// ---------------- stage 4: x-irFFT as GEMM with weighted tables ----------------
__global__ void __launch_bounds__(32) k_s4(float* __restrict__ ws) {
  const int lane = threadIdx.x, lm = lane & 15, hi = lane >> 4;
  const int m0 = blockIdx.x * 16;                  // y rows
  const int n0 = blockIdx.y * 16;                  // x cols
  const int b  = blockIdx.z;
  const size_t pb = (size_t)b * 36864 + (size_t)(m0 + lm) * 144 + 2 * hi;
  const float* Aqr = ws + OFF_PQR + pb;
  const float* Aqi = ws + OFF_PQI + pb;
  const float* Avr = ws + OFF_PVR + pb;
  const float* Avi = ws + OFF_PVI + pb;
  const float* Bc = ws + OFF_WC  + (size_t)(n0 + lm) * 144 + 2 * hi;
  const float* Bs = ws + OFF_WS2 + (size_t)(n0 + lm) * 144 + 2 * hi;
  v8f c0 = {0.f,0.f,0.f,0.f,0.f,0.f,0.f,0.f}, c1 = c0, c2 = c0, c3 = c0;
  for (int k = 0; k < 144; k += 4) {
    v2f bc = *(const v2f*)(Bc + k);
    v2f bs = *(const v2f*)(Bs + k);
    v2f ar = *(const v2f*)(Aqr + k);
    v2f ai = *(const v2f*)(Aqi + k);
    v2f wr = *(const v2f*)(Avr + k);
    v2f wi = *(const v2f*)(Avi + k);
    c0 = wmma4(c0, ar, bc);
    c1 = wmma4(c1, ai, bs);
    c2 = wmma4(c2, wr, bc);
    c3 = wmma4(c3, wi, bs);
  }
  float* QF = ws + OFF_QF + (size_t)b * NN * NN;
  float* VF = ws + OFF_VF + (size_t)b * NN * NN;
  const int mrow = m0 + 8 * hi;
  for (int r = 0; r < 8; ++r) {
    QF[(size_t)(mrow + r) * 256 + n0 + lm] = c0[r] - c1[r];
    VF[(size_t)(mrow + r) * 256 + n0 + lm] = c2[r] - c3[r];
  }
}

// ---------------- per-batch plane reductions: nn half ----------------
__global__ void k_reduce_nn(const float* __restrict__ u, float* __restrict__ ws) {
  __shared__ float sm[256];
  const int t = threadIdx.x;
  const int b = blockIdx.x;
  const float* up = u + (size_t)b * NN * NN * 2;
  float S0 = 0, S1 = 0, Q = 0, Cx = 0, Cy = 0;
  for (int e = t; e < NN * NN; e += 256) {
    int y = e >> 8, x = e & 255;
    __builtin_prefetch(up + ((size_t)((y + 2) & 255) * NN + x) * 2, 0, 1);
    float u0  = up[((size_t)y * NN + x) * 2];
    float u1  = up[((size_t)y * NN + x) * 2 + 1];
    float uxp = up[((size_t)y * NN + ((x + 1) & 255)) * 2];
    float uyp = up[((size_t)((y + 1) & 255) * NN + x) * 2];
    S0 += u0; S1 += u1; Q += u0 * u0; Cx += u0 * uxp; Cy += u0 * uyp;
  }
  float vals[5] = {S0, S1, Q, Cx, Cy};
  for (int k = 0; k < 5; ++k) {
    sm[t] = vals[k]; __syncthreads();
    for (int s = 128; s > 0; s >>= 1) { if (t < s) sm[t] += sm[t + s]; __syncthreads(); }
    if (t == 0) ws[OFF_NNS + (size_t)b * 8 + k] = sm[0];
    __syncthreads();
  }
}

// ---------------- per-batch plane reductions: ns half (over q,v) ----------------
__global__ void k_reduce_ns(float* __restrict__ ws) {
  __shared__ float sm[256];
  const int t = threadIdx.x;
  const int b = blockIdx.x;
  const float* qf = ws + OFF_QF + (size_t)b * NN * NN;
  const float* vf = ws + OFF_VF + (size_t)b * NN * NN;
  float Au = 0, Bv = 0, QV = 0, Cxq = 0, Cxv = 0, Cyv = 0, P1 = 0, P2 = 0, P3 = 0, P4 = 0;
  for (int e = t; e < NN * NN; e += 256) {
    int y = e >> 8, x = e & 255;
    int xp = (x + 1) & 255, yp = (y + 1) & 255;
    float q   = qf[(size_t)y * NN + x],  v   = vf[(size_t)y * NN + x];
    float qxp = qf[(size_t)y * NN + xp], vxp = vf[(size_t)y * NN + xp];
    float qyp = qf[(size_t)yp * NN + x], vyp = vf[(size_t)yp * NN + x];
    Au += q * q; Bv += v * v; QV += q * v;
    Cxq += q * qxp; Cxv += v * vxp; Cyv += v * vyp;
    P1 += v * qyp; P2 += q * vyp; P3 += q * vxp; P4 += v * qxp;
  }
  float vals[10] = {Au, Bv, QV, Cxq, Cxv, Cyv, P1, P2, P3, P4};
  for (int k = 0; k < 10; ++k) {
    sm[t] = vals[k]; __syncthreads();
    for (int s = 128; s > 0; s >>= 1) { if (t < s) sm[t] += sm[t + s]; __syncthreads(); }
    if (t == 0) ws[OFF_NSS + (size_t)b * 16 + k] = sm[0];
    __syncthreads();
  }
}

// ---------------- build xi rows (128 x [ua, ud, flag, norm]) ----------------
__global__ void k_build_xi(const float* __restrict__ vs, const float* __restrict__ dt,
                           const float* __restrict__ dxp, float* __restrict__ ws) {
  int i = threadIdx.x;
  if (i >= 128) return;
  float dxv = dxp[0];
  float ua, ud, fl;
  if (i < 64) {
    const float* S = ws + OFF_NNS;
    float S0 = S[i*8+0], S1 = S[i*8+1], Q = S[i*8+2], Cx = S[i*8+3], Cy = S[i*8+4];
    float S0m = S[((i + 63) & 63) * 8 + 0];
    float S0p = S[((i + 1) & 63) * 8 + 0];
    float c0 = vs[i*4+0], cx = vs[i*4+1], cy = vs[i*4+2];
    float dti = dt[i];
    float rd = (S1 - S0) - c0 * dti * (S0m + S0p - 2.f * S0) / (dxv * dxv);
    ud = c0 * expf(-rd * rd);
    float ra = (S1 - S0) + dti * (fabsf(cx) * (Q - Cx) + fabsf(cy) * (Q - Cy)) / dxv;
    ua = sqrtf(cx * cx + cy * cy) * expf(-ra * ra);
    fl = 0.f;
  } else {
    int bn = i - 64;
    const float* S = ws + OFF_NSS + (size_t)bn * 16;
    float Au = S[0], Bv = S[1], QV = S[2], Cxq = S[3], Cxv = S[4], Cyv = S[5];
    float P1 = S[6], P2 = S[7], P3 = S[8], P4 = S[9];
    float c0 = vs[i*4+0], cx = vs[i*4+1], cy = vs[i*4+2];
    float dti = dt[i];
    float sau, sav;
    if      (cx <= 0.f && cy >= 0.f) sau = -cx * (Au - Cxq) + cy * (QV - P2);
    else if (cx >= 0.f && cy >= 0.f) sau =  cx * (Au - Cxq) + cy * (QV - P2);
    else if (cx <= 0.f && cy <= 0.f) sau = -cx * (Au - Cxq) - cy * (QV - P1);
    else                             sau =  cx * (Au - Cxq) - cy * (QV - P1);
    if      (cy <= 0.f && cx >= 0.f) sav = -cy * (Bv - Cxv) + cx * (QV - P1);
    else if (cy >= 0.f && cx >= 0.f) sav =  cy * (QV - P4) + cx * (Bv - Cyv);
    else if (cy <= 0.f && cx <= 0.f) sav = -cy * (QV - P3) - cx * (Bv - Cyv);
    else                             sav =  cy * (QV - P4) - cx * (Bv - Cyv);
    float tu = dti * sau / dxv;
    float tv = dti * sav / dxv;
    float ru = cx * expf(-tu * tu);
    float rv = cy * expf(-tv * tv);
    ua = sqrtf(ru * ru + rv * rv);
    ud = c0;
    fl = 1.f;
  }
  ws[OFF_XI + i*4+0] = ua;
  ws[OFF_XI + i*4+1] = ud;
  ws[OFF_XI + i*4+2] = fl;
  ws[OFF_XI + i*4+3] = sqrtf(ua * ua + ud * ud + fl * fl);
}

// ---------------- row-normalize reps = concat(x2, x1) ----------------
__global__ void k_normalize(const float* __restrict__ x1, const float* __restrict__ x2,
                            float* __restrict__ ws) {
  __shared__ float sm[256];
  int t = threadIdx.x, r = blockIdx.x;
  const float* src = (r < 128) ? (x2 + (size_t)r * DDIM) : (x1 + (size_t)(r - 128) * DDIM);
  float a0 = src[t], a1 = src[t + 256];
  sm[t] = a0 * a0 + a1 * a1; __syncthreads();
  for (int s = 128; s > 0; s >>= 1) { if (t < s) sm[t] += sm[t + s]; __syncthreads(); }
  float nrm = fmaxf(sqrtf(sm[0]), 1e-8f);
  ws[OFF_RN + (size_t)r * DDIM + t]       = a0 / nrm;
  ws[OFF_RN + (size_t)r * DDIM + t + 256] = a1 / nrm;
}

// ---------------- smat = rn * rn^T via V_WMMA_F32_16X16X4_F32 ----------------
__global__ void __launch_bounds__(32) k_gemm_wmma(float* __restrict__ ws) {
  const float* rn = ws + OFF_RN;
  float* smat = ws + OFF_SMAT;
  const int lane = threadIdx.x, lm = lane & 15, hi = lane >> 4;
  const int m0 = blockIdx.x * 16, n0 = blockIdx.y * 16;
  const float* arow = rn + (size_t)(m0 + lm) * DDIM + 2 * hi;
  const float* brow = rn + (size_t)(n0 + lm) * DDIM + 2 * hi;
  v8f c = {0.f,0.f,0.f,0.f,0.f,0.f,0.f,0.f};
  for (int k = 0; k < DDIM; k += 4) {
    v2f a = *(const v2f*)(arow + k);
    v2f b = *(const v2f*)(brow + k);
    c = wmma4(c, a, b);
  }
  const int rbase = m0 + 8 * hi;
  for (int r = 0; r < 8; ++r)
    smat[(size_t)(rbase + r) * TWOB + n0 + lm] = c[r];
}

// ---------------- per-row NT-Xent with PDE-similarity weights ----------------
__global__ void k_rowloss(float* __restrict__ ws) {
  __shared__ float sm[256];
  const int J = threadIdx.x, I = blockIdx.x;
  const float* smat = ws + OFF_SMAT;
  const float* xi = ws + OFF_XI;
  const int posJ = (I + 128) & 255;
  const float pos = smat[(size_t)I * TWOB + posJ] / kTemp;
  float val = -INFINITY;
  if (J != I && J != posJ) {
    int ii = I & 127, jj = J & 127;
    float ai = xi[ii*4], di = xi[ii*4+1], fi = xi[ii*4+2], ni = xi[ii*4+3];
    float aj = xi[jj*4], dj = xi[jj*4+1], fj = xi[jj*4+2], nj = xi[jj*4+3];
    float prod = sqrtf(fabsf(ai * aj) + fabsf(di * dj) + fabsf(fi * fj));
    float nm = fmaxf(ni, nj);
    float w = 0.f;
    if (nm > 0.f) {
      float s = 1.f - kLam * (prod / nm);
      w = __builtin_isfinite(s) ? s : 0.f;      // nan_to_num
    }
    val = w * smat[(size_t)I * TWOB + J] / kTemp;
  }
  sm[J] = val; __syncthreads();
  for (int s = 128; s > 0; s >>= 1) { if (J < s) sm[J] = fmaxf(sm[J], sm[J + s]); __syncthreads(); }
  float m = fmaxf(sm[0], pos); __syncthreads();
  sm[J] = (val == -INFINITY) ? 0.f : expf(val - m); __syncthreads();
  for (int s = 128; s > 0; s >>= 1) { if (J < s) sm[J] += sm[J + s]; __syncthreads(); }
  if (J == 0) {
    float lse = m + logf(sm[0] + expf(pos - m));
    ws[OFF_RL + I] = -(pos - lse);
  }
}

__global__ void k_final(const float* __restrict__ ws, float* __restrict__ out) {
  __shared__ float sm[256];
  int t = threadIdx.x;
  sm[t] = ws[OFF_RL + t]; __syncthreads();
  for (int s = 128; s > 0; s >>= 1) { if (t < s) sm[t] += sm[t + s]; __syncthreads(); }
  if (t == 0) out[0] = sm[0] / 256.0f;
}

extern "C" void kernel_launch(void* const* d_in, const int* in_sizes, int n_in,
                              void* d_out, int out_size, void* d_ws, size_t ws_size,
                              hipStream_t stream) {
  const float* x1 = (const float*)d_in[0];
  const float* x2 = (const float*)d_in[1];
  const float* vs = (const float*)d_in[2];
  const float* u  = (const float*)d_in[3];
  const float* dt = (const float*)d_in[4];
  const float* dx = (const float*)d_in[5];
  float* ws  = (float*)d_ws;
  float* out = (float*)d_out;

  k_init_tw<<<256, 256, 0, stream>>>(ws);
  k_tr<<<dim3(8, 8, 64), dim3(32, 8), 0, stream>>>(u, ws);
  k_s1<<<dim3(16, 16, 64), 32, 0, stream>>>(ws);
  k_s2<<<dim3(9, 16, 64), 32, 0, stream>>>(ws);
  k_s3<<<dim3(16, 9, 64), 32, 0, stream>>>(ws);
  k_s4<<<dim3(16, 16, 64), 32, 0, stream>>>(ws);
  k_reduce_nn<<<64, 256, 0, stream>>>(u, ws);
  k_reduce_ns<<<64, 256, 0, stream>>>(ws);
  k_build_xi<<<1, 128, 0, stream>>>(vs, dt, dx, ws);
  k_normalize<<<256, 256, 0, stream>>>(x1, x2, ws);
  k_gemm_wmma<<<dim3(16, 16), 32, 0, stream>>>(ws);
  k_rowloss<<<256, 256, 0, stream>>>(ws);
  k_final<<<1, 256, 0, stream>>>(ws, out);
}